// AttentionMemory_86517821216032
// MI455X (gfx1250) — compile-verified
//
#include <hip/hip_runtime.h>
#include <hip/hip_bf16.h>
#include <math.h>

typedef __attribute__((ext_vector_type(2))) float v2f;
typedef __attribute__((ext_vector_type(8))) float v8f;

#define HW   4096
#define CK   64
#define NT   64          // n-columns per workgroup
#define MT   64          // m-rows per iteration
#define LDSP 68          // LDS row stride in floats (272B, 16B aligned)

__device__ __forceinline__ unsigned fmap(float f) {
    // monotone float -> uint map so atomicMax(uint) orders floats
    unsigned u = __float_as_uint(f);
    return (u & 0x80000000u) ? ~u : (u | 0x80000000u);
}
__device__ __forceinline__ float funmap(unsigned u) {
    unsigned b = (u & 0x80000000u) ? (u & 0x7FFFFFFFu) : ~u;
    return __uint_as_float(b);
}

// Issue one 64c x 64m Mk tile as per-lane async B128 copies into LDS.
// 256 threads x 4 issues x 16B = 16 KB; tracked by ASYNCcnt.
__device__ __forceinline__ void mk_tile_async(float* ldsBuf, const float* gbase, int tid) {
    #pragma unroll
    for (int i = 0; i < 4; ++i) {
        int idx = tid + 256 * i;            // 0..1023 quads
        int c   = idx >> 4;                 // 0..63 (channel row)
        int q   = idx & 15;                 // 0..15 (float4 within row)
        unsigned lds_addr = (unsigned)(uintptr_t)(ldsBuf + c * LDSP + q * 4);
        const float* ga = gbase + (size_t)c * HW + q * 4;
        asm volatile("global_load_async_to_lds_b128 %0, %1, off"
                     :: "v"(lds_addr), "v"(ga)
                     : "memory");
    }
}

__device__ __forceinline__ void wait_async0() {
    asm volatile("s_wait_asynccnt 0" ::: "memory");
}

__global__ __launch_bounds__(256)
void attn_mem_softmax_kernel(const float* __restrict__ Mk,
                             const float* __restrict__ Qk,
                             float* __restrict__ out) {
    __shared__ float ldsMk[2][CK * LDSP];   // double-buffered [c][m] tiles
    __shared__ float ldsQk[CK * LDSP];      // [c][n] tile, pre-scaled by 1/8
    __shared__ unsigned colMaxU[NT];
    __shared__ float colMax[NT];
    __shared__ float colSum[NT];
    __shared__ float colInv[NT];

    const int tid  = threadIdx.x;
    const int lane = tid & 31;
    const int wave = tid >> 5;              // 0..7
    const int b    = blockIdx.y;            // 0..7
    const int n0   = blockIdx.x * NT;       // 0..4032

    const float* Mkb = Mk + (size_t)b * CK * HW;
    const float* Qkb = Qk + (size_t)b * CK * HW;
    const size_t outBase = (size_t)b * HW * HW;

    // kick off async fill of the first Mk tile while we stage Qk
    mk_tile_async(ldsMk[0], Mkb, tid);

    // ---- load Qk tile (64c x 64n), scale by 1/sqrt(CK) ----
    #pragma unroll
    for (int i = 0; i < 4; ++i) {
        int idx = tid + 256 * i;            // 0..1023 float4 quads
        int c   = idx >> 4;
        int q   = idx & 15;
        float4 v = *reinterpret_cast<const float4*>(Qkb + (size_t)c * HW + n0 + q * 4);
        v.x *= 0.125f; v.y *= 0.125f; v.z *= 0.125f; v.w *= 0.125f;
        *reinterpret_cast<float4*>(&ldsQk[c * LDSP + q * 4]) = v;
    }
    if (tid < NT) { colMaxU[tid] = 0u; }    // maps below any finite float

    // per-wave tile assignment: 4 m-tiles x 4 n-tiles = 16 tiles, 2 per wave
    const int mt  = wave & 3;               // m-tile 0..3
    const int nt0 = (wave >> 2) * 2;        // n-tile pair base: 0 or 2
    const int am  = lane & 15;              // A row within tile / B,C col
    const int kh  = (lane >> 4) * 2;        // K offset of VGPR0 per half-wave
    const int mcol  = mt * 16 + am;         // column into ldsMk
    const int b0col = nt0 * 16 + am;        // column into ldsQk (tile nt0)
    const int b1col = b0col + 16;           // tile nt0+1
    const int rhalf = (lane >> 4) << 3;     // C/D rows: 0..7 or 8..15

    // ---- phase 1: S = Mk^T Qk, write raw scores, track column max ----
    for (int it = 0; it < HW / MT; ++it) {
        const int m0 = it * MT;
        wait_async0();                      // own async copies landed
        __syncthreads();                    // everyone's landed; prior readers done
        if (it + 1 < HW / MT)               // prefetch next tile into other buffer
            mk_tile_async(ldsMk[(it + 1) & 1], Mkb + (m0 + MT), tid);

        const float* mkT = ldsMk[it & 1];
        v8f acc0 = {};
        v8f acc1 = {};
        #pragma unroll
        for (int kk = 0; kk < 16; ++kk) {
            int kb = kk * 4 + kh;
            v2f a, p, q;
            a.x = mkT[(kb    ) * LDSP + mcol];
            a.y = mkT[(kb + 1) * LDSP + mcol];
            p.x = ldsQk[(kb    ) * LDSP + b0col];
            p.y = ldsQk[(kb + 1) * LDSP + b0col];
            q.x = ldsQk[(kb    ) * LDSP + b1col];
            q.y = ldsQk[(kb + 1) * LDSP + b1col];
            acc0 = __builtin_amdgcn_wmma_f32_16x16x4_f32(false, a, false, p,
                                                         (short)0, acc0, false, false);
            acc1 = __builtin_amdgcn_wmma_f32_16x16x4_f32(false, a, false, q,
                                                         (short)0, acc1, false, false);
        }

        // write raw scores + update per-column running max
        const int mrow = m0 + mt * 16 + rhalf;
        const int c0   = n0 + b0col;
        const int c1   = n0 + b1col;
        float mx0 = acc0[0], mx1 = acc1[0];
        #pragma unroll
        for (int r = 0; r < 8; ++r) {
            float s0 = acc0[r], s1 = acc1[r];
            size_t row = outBase + (size_t)(mrow + r) * HW;
            out[row + c0] = s0;
            out[row + c1] = s1;
            mx0 = fmaxf(mx0, s0);
            mx1 = fmaxf(mx1, s1);
        }
        atomicMax(&colMaxU[b0col], fmap(mx0));
        atomicMax(&colMaxU[b1col], fmap(mx1));
    }

    __threadfence();                        // raw scores visible to whole WG
    __syncthreads();
    if (tid < NT) { colMax[tid] = funmap(colMaxU[tid]); colSum[tid] = 0.f; }
    __syncthreads();

    // ---- phase 2: column sums of exp(s - max), stripe stays in L2 ----
    const int nn = tid & 63;
    const int ms = tid >> 6;                // 4 threads per column
    const float cm = colMax[nn];
    float* stripe = out + outBase + n0 + nn;
    float psum = 0.f;
    for (int m = ms; m < HW; m += 4)
        psum += expf(stripe[(size_t)m * HW] - cm);
    atomicAdd(&colSum[nn], psum);
    __syncthreads();
    if (tid < NT) colInv[tid] = 1.0f / colSum[tid];
    __syncthreads();

    // ---- phase 3: normalize in place ----
    const float inv = colInv[nn];
    for (int m = ms; m < HW; m += 4) {
        size_t off = (size_t)m * HW;
        stripe[off] = expf(stripe[off] - cm) * inv;
    }
}

extern "C" void kernel_launch(void* const* d_in, const int* in_sizes, int n_in,
                              void* d_out, int out_size, void* d_ws, size_t ws_size,
                              hipStream_t stream) {
    (void)in_sizes; (void)n_in; (void)out_size; (void)d_ws; (void)ws_size;
    const float* Mk = (const float*)d_in[0];
    const float* Qk = (const float*)d_in[1];
    float* out = (float*)d_out;
    dim3 grid(HW / NT, 8);                  // 64 column-stripes x 8 batches
    dim3 block(256);                        // 8 wave32s
    attn_mem_softmax_kernel<<<grid, block, 0, stream>>>(Mk, Qk, out);
}